// ContrastiveLoss_79035988181264
// MI455X (gfx1250) — compile-verified
//
#include <hip/hip_runtime.h>
#include <math.h>

typedef __attribute__((ext_vector_type(2))) float v2f;
typedef __attribute__((ext_vector_type(4))) float v4f;
typedef __attribute__((ext_vector_type(8))) float v8f;

#define EPSN 1e-8f

static constexpr int B = 16, T = 4096, N = 512, D = 512;

// ---------------------------------------------------------------------------
// Kernel A: per-row  n_c = max(||c||,eps),  q_t = exp(dot(y_t,c)/(n_t*n_c))
// one wave32 per row, float4 loads, shuffle reduction
// ---------------------------------------------------------------------------
__global__ __launch_bounds__(256)
void row_qt_kernel(const float* __restrict__ c, const float* __restrict__ y,
                   float* __restrict__ n_c, float* __restrict__ q_t)
{
  const int lane = threadIdx.x & 31;
  const int row  = blockIdx.x * (blockDim.x >> 5) + (threadIdx.x >> 5);
  if (row >= B * T) return;
  const float* cr = c + (size_t)row * D;
  const float* yr = y + (size_t)row * D;
  float scc = 0.f, syy = 0.f, syc = 0.f;
  #pragma unroll
  for (int k = lane * 4; k < D; k += 128) {
    v4f cv = *(const v4f*)(cr + k);
    v4f yv = *(const v4f*)(yr + k);
    scc += cv.x*cv.x + cv.y*cv.y + cv.z*cv.z + cv.w*cv.w;
    syy += yv.x*yv.x + yv.y*yv.y + yv.z*yv.z + yv.w*yv.w;
    syc += yv.x*cv.x + yv.y*cv.y + yv.z*cv.z + yv.w*cv.w;
  }
  #pragma unroll
  for (int m = 16; m; m >>= 1) {
    scc += __shfl_xor(scc, m, 32);
    syy += __shfl_xor(syy, m, 32);
    syc += __shfl_xor(syc, m, 32);
  }
  if (lane == 0) {
    float nc = fmaxf(sqrtf(scc), EPSN);
    float nt = fmaxf(sqrtf(syy), EPSN);
    n_c[row] = nc;
    q_t[row] = __expf(syc / (nt * nc));
  }
}

// ---------------------------------------------------------------------------
// Kernel B: n_d = max(||y_distraction||, eps) per row
// ---------------------------------------------------------------------------
__global__ __launch_bounds__(256)
void row_nd_kernel(const float* __restrict__ yd, float* __restrict__ n_d)
{
  const int lane = threadIdx.x & 31;
  const int row  = blockIdx.x * (blockDim.x >> 5) + (threadIdx.x >> 5);
  if (row >= B * N) return;
  const float* r = yd + (size_t)row * D;
  float s = 0.f;
  #pragma unroll
  for (int k = lane * 4; k < D; k += 128) {
    v4f v = *(const v4f*)(r + k);
    s += v.x*v.x + v.y*v.y + v.z*v.z + v.w*v.w;
  }
  #pragma unroll
  for (int m = 16; m; m >>= 1) s += __shfl_xor(s, m, 32);
  if (lane == 0) n_d[row] = fmaxf(sqrtf(s), EPSN);
}

// ---------------------------------------------------------------------------
// Kernel C: fused fp32 WMMA GEMM (c x y_d^T per batch) + exp/row-sum epilogue
// + final loss reduction.
// Workgroup = 8 waves = 128 t-rows x 64 n-cols tile; each wave owns a 32x32
// output tile (2x2 fragments -> 4 WMMAs per 4 DS loads). Scores never touch
// global memory.
// ---------------------------------------------------------------------------
__global__ __launch_bounds__(256)
void gemm_loss_kernel(const float* __restrict__ c,  const float* __restrict__ yd,
                      const float* __restrict__ n_c, const float* __restrict__ n_d,
                      const float* __restrict__ q_t, float* __restrict__ out)
{
  constexpr int LDA = 68;                 // 64 + 4 pad: conflict-free fragment reads
  __shared__ float As[128 * LDA];
  __shared__ float Bs[64 * LDA];
  __shared__ float rowsum[128];
  __shared__ float invnc[128];

  const int b    = blockIdx.y;
  const int t0   = blockIdx.x * 128;
  const int tid  = threadIdx.x;
  const int lane = tid & 31;
  const int wave = tid >> 5;
  const int mw   = wave >> 1;             // 0..3 : which 32-row band
  const int nw   = wave & 1;              // 0..1 : which 32-col band
  const int l15  = lane & 15;
  const int kh   = lane >> 4;             // K-pair select within fragment

  if (tid < 128) {
    rowsum[tid] = 0.f;
    invnc[tid]  = 1.f / n_c[(size_t)b * T + t0 + tid];
  }
  __syncthreads();

  const float* Cb = c  + (size_t)(b * T + t0) * D;
  const float* Db = yd + (size_t)b * N * D;

  for (int n0 = 0; n0 < N; n0 += 64) {
    v8f acc00 = {}, acc01 = {}, acc10 = {}, acc11 = {};
    for (int k0 = 0; k0 < D; k0 += 64) {
      __syncthreads();                    // previous tile fully consumed
      // stage A: 128x64 (8 b128 per thread), B: 64x64 (4 b128 per thread)
      #pragma unroll
      for (int i = 0; i < 8; ++i) {
        int idx = tid + 256 * i;          // 0..2047
        int row = idx >> 4;
        int q4  = (idx & 15) * 4;
        *(v4f*)&As[row * LDA + q4] = *(const v4f*)(Cb + (size_t)row * D + k0 + q4);
      }
      #pragma unroll
      for (int i = 0; i < 4; ++i) {
        int idx = tid + 256 * i;          // 0..1023
        int row = idx >> 4;
        int q4  = (idx & 15) * 4;
        *(v4f*)&Bs[row * LDA + q4] = *(const v4f*)(Db + (size_t)(n0 + row) * D + k0 + q4);
      }
      // prefetch next K-chunk toward WGP while we compute this one
      if (k0 + 64 < D) {
        __builtin_prefetch(Cb + (size_t)(tid >> 1) * D + k0 + 64 + (tid & 1) * 32, 0, 0);
        if (tid < 128)
          __builtin_prefetch(Db + (size_t)(n0 + (tid >> 1)) * D + k0 + 64 + (tid & 1) * 32, 0, 0);
      }
      __syncthreads();
      const float* Ap0 = &As[(mw * 32 + l15) * LDA + 2 * kh];
      const float* Ap1 = Ap0 + 16 * LDA;
      const float* Bp0 = &Bs[(nw * 32 + l15) * LDA + 2 * kh];
      const float* Bp1 = Bp0 + 16 * LDA;
      #pragma unroll
      for (int kk = 0; kk < 64; kk += 4) {
        v2f a0 = *(const v2f*)(Ap0 + kk);
        v2f a1 = *(const v2f*)(Ap1 + kk);
        v2f b0 = *(const v2f*)(Bp0 + kk);
        v2f b1 = *(const v2f*)(Bp1 + kk);
        acc00 = __builtin_amdgcn_wmma_f32_16x16x4_f32(false, a0, false, b0,
                                                      (short)0, acc00, false, false);
        acc01 = __builtin_amdgcn_wmma_f32_16x16x4_f32(false, a0, false, b1,
                                                      (short)0, acc01, false, false);
        acc10 = __builtin_amdgcn_wmma_f32_16x16x4_f32(false, a1, false, b0,
                                                      (short)0, acc10, false, false);
        acc11 = __builtin_amdgcn_wmma_f32_16x16x4_f32(false, a1, false, b1,
                                                      (short)0, acc11, false, false);
      }
    }
    // epilogue: scale by 1/(n_c*n_d), exp, reduce lanes per row, LDS-accumulate
    const int   colb   = n0 + nw * 32;
    const float invnd0 = 1.f / n_d[(size_t)b * N + colb + l15];
    const float invnd1 = 1.f / n_d[(size_t)b * N + colb + 16 + l15];
    #pragma unroll
    for (int r = 0; r < 8; ++r) {
      int row0 = mw * 32 + r + 8 * kh;    // rows of frag-0 element r in this lane
      int row1 = row0 + 16;               // rows of frag-1
      float inc0 = invnc[row0], inc1 = invnc[row1];
      // two N-frags of the same row summed in-register first
      float s0 = __expf(acc00[r] * inc0 * invnd0) + __expf(acc01[r] * inc0 * invnd1);
      float s1 = __expf(acc10[r] * inc1 * invnd0) + __expf(acc11[r] * inc1 * invnd1);
      #pragma unroll
      for (int m = 8; m; m >>= 1) {
        s0 += __shfl_xor(s0, m, 32);
        s1 += __shfl_xor(s1, m, 32);
      }
      if (l15 == 0) {
        atomicAdd(&rowsum[row0], s0);
        atomicAdd(&rowsum[row1], s1);
      }
    }
  }
  __syncthreads();
  // loss contribution: log(rowsum + q_t) - log(q_t), reduced and atomically added
  if (tid < 128) {
    float qt = q_t[(size_t)b * T + t0 + tid];
    float v  = __logf(rowsum[tid] + qt) - __logf(qt);
    #pragma unroll
    for (int m = 16; m; m >>= 1) v += __shfl_xor(v, m, 32);
    if (lane == 0) atomicAdd(out, v);
  }
}

// ---------------------------------------------------------------------------
extern "C" void kernel_launch(void* const* d_in, const int* in_sizes, int n_in,
                              void* d_out, int out_size, void* d_ws, size_t ws_size,
                              hipStream_t stream)
{
  const float* c  = (const float*)d_in[0];   // [B,T,D]
  const float* yt = (const float*)d_in[1];   // [B,T,D]
  const float* yd = (const float*)d_in[2];   // [B,N,D]
  float* out = (float*)d_out;                // scalar loss

  float* n_c = (float*)d_ws;                 // B*T floats
  float* q_t = n_c + (size_t)B * T;          // B*T floats
  float* n_d = q_t + (size_t)B * T;          // B*N floats

  hipMemsetAsync(out, 0, sizeof(float), stream);

  row_qt_kernel<<<dim3((B * T) / 8), dim3(256), 0, stream>>>(c, yt, n_c, q_t);
  row_nd_kernel<<<dim3((B * N) / 8), dim3(256), 0, stream>>>(yd, n_d);
  gemm_loss_kernel<<<dim3(T / 128, B), dim3(256), 0, stream>>>(c, yd, n_c, n_d, q_t, out);
}